// ChannelSubgraphLayer_64355789963650
// MI455X (gfx1250) — compile-verified
//
#include <hip/hip_runtime.h>
#include <cstdint>

#define GAT_IN    128
#define GAT_HID   256
#define GAT_HEADS 4
#define GAT_C     64
#define NEG_SLOPE 0.2f

// Flip to 0 if the async-to-LDS mnemonic is rejected by the assembler.
#define USE_ASYNC_LDS 1

typedef float v2f __attribute__((ext_vector_type(2)));
typedef float v8f __attribute__((ext_vector_type(8)));

#ifdef __has_builtin
#if __has_builtin(__builtin_amdgcn_wmma_f32_16x16x4_f32)
#define HAVE_WMMA_F32 1
#endif
#endif
#ifndef HAVE_WMMA_F32
#define HAVE_WMMA_F32 0
#endif

// ---------------------------------------------------------------------------
// helpers
// ---------------------------------------------------------------------------
__device__ __forceinline__ unsigned f32_flip(float f) {
  unsigned u = __float_as_uint(f);
  return u ^ ((unsigned)((int)u >> 31) | 0x80000000u);
}
__device__ __forceinline__ float f32_unflip(unsigned u) {
  unsigned m = (u & 0x80000000u) ? 0x80000000u : 0xFFFFFFFFu;
  return __uint_as_float(u ^ m);
}

__device__ __forceinline__ void edge_sd(const int* __restrict__ ei, int E, int e,
                                        int& s, int& d) {
  if (e < E) { s = ei[e]; d = ei[E + e]; }
  else       { s = e - E; d = e - E; }      // appended self-loops
}

// generic LDS pointer -> raw LDS byte address (low 32 bits of the flat addr)
__device__ __forceinline__ unsigned lds_addr_of(const void* p) {
  return (unsigned)(reinterpret_cast<uintptr_t>(p));
}

__device__ __forceinline__ void wait_asynccnt0() {
#if defined(__has_builtin) && __has_builtin(__builtin_amdgcn_s_wait_asynccnt)
  __builtin_amdgcn_s_wait_asynccnt(0);
#else
  asm volatile("s_wait_asynccnt 0" ::: "memory");
#endif
}

// ---------------------------------------------------------------------------
// zero-init kernel (agg + emax + denom are laid out contiguously)
// ---------------------------------------------------------------------------
__global__ void zero_u32_kernel(unsigned* __restrict__ p, int count) {
  int i = blockIdx.x * blockDim.x + threadIdx.x;
  if (i < count) p[i] = 0u;
}

// ---------------------------------------------------------------------------
// GEMM: out[M, 256] = X[M, K] @ W[K, 256] (+ bias) using V_WMMA_F32_16X16X4_F32.
// Block = 8 waves -> 128 rows x 64 cols. The shared B tile (K x 64, 32-64 KB)
// is staged once per block into LDS with GLOBAL_LOAD_ASYNC_TO_LDS_B128
// (ASYNCcnt), then all 8 waves feed their WMMA B fragments from LDS.
// A frag (16x4 f32): lanes 0-15 hold K=0,1 / lanes 16-31 K=2,3 (ISA 7.12.2)
// C/D frag: VGPR i -> row m0+i (lanes 0-15) / m0+8+i (lanes 16-31), col=lane&15
// ---------------------------------------------------------------------------
template<int K>
__global__ __launch_bounds__(256)
void wmma_gemm_kernel(const float* __restrict__ X, const float* __restrict__ W,
                      const float* __restrict__ bias, float* __restrict__ out, int M)
{
  constexpr int NC = GAT_HID;
  __shared__ float Btile[K * 64];            // [k][64] row-major

  const int lane = threadIdx.x & 31;
  const int wave = threadIdx.x >> 5;
  const int m0 = (blockIdx.x * 8 + wave) * 16;
  const int n0 = blockIdx.y * 64;

  // ---- cooperative B-tile staging (all 256 threads, float4 granules) ----
  for (int f4 = threadIdx.x * 4; f4 < K * 64; f4 += 256 * 4) {
    const int kk = f4 >> 6;                  // B row (k index)
    const int cc = f4 & 63;                  // B col within tile
    const float* gp = W + (size_t)kk * NC + n0 + cc;
#if USE_ASYNC_LDS
    const unsigned la = lds_addr_of(&Btile[f4]);
    asm volatile("global_load_async_to_lds_b128 %0, %1, off"
                 :: "v"(la), "v"(gp) : "memory");
#else
    *(float4*)&Btile[f4] = *(const float4*)gp;
#endif
  }
#if USE_ASYNC_LDS
  wait_asynccnt0();                          // own wave's async copies done
#endif
  __syncthreads();                           // tile visible to all waves

  if (m0 >= M) return;                       // wave-uniform: EXEC stays all-ones

  const int arow = m0 + (lane & 15);
  const int kl   = (lane >> 4) << 1;         // 0 or 2
  const int bcol = lane & 15;

  v8f acc[4] = {};
#if HAVE_WMMA_F32
  for (int k = 0; k < K; k += 4) {
    v2f a;
    a.x = X[(size_t)arow * K + k + kl];
    a.y = X[(size_t)arow * K + k + kl + 1];
#pragma unroll
    for (int t = 0; t < 4; ++t) {
      v2f b;
      b.x = Btile[(k + kl) * 64 + t * 16 + bcol];
      b.y = Btile[(k + kl + 1) * 64 + t * 16 + bcol];
      acc[t] = __builtin_amdgcn_wmma_f32_16x16x4_f32(
          false, a, false, b, (short)0, acc[t], false, false);
    }
  }
#else
  // scalar fallback with identical output mapping
  const int srow0 = m0 + ((lane >> 4) << 3);
#pragma unroll
  for (int t = 0; t < 4; ++t) {
    const int c = t * 16 + bcol;
#pragma unroll
    for (int i = 0; i < 8; ++i) {
      float s = 0.f;
      for (int k = 0; k < K; ++k)
        s += X[(size_t)(srow0 + i) * K + k] * Btile[k * 64 + c];
      acc[t][i] = s;
    }
  }
#endif
  const int srow = m0 + ((lane >> 4) << 3);
#pragma unroll
  for (int t = 0; t < 4; ++t) {
    const int c = n0 + t * 16 + bcol;
    const float bv = bias ? bias[c] : 0.0f;
#pragma unroll
    for (int i = 0; i < 8; ++i)
      out[(size_t)(srow + i) * NC + c] = acc[t][i] + bv;
  }
}

// ---------------------------------------------------------------------------
// Per-node attention logits: e_src[n,h] = <hx[n,h,:], a_src[h,:]>, same for dst.
// One wave per node; lane covers channels lane*8..lane*8+7; head = lane>>3.
// ---------------------------------------------------------------------------
__global__ __launch_bounds__(256)
void attn_logits_kernel(const float* __restrict__ hx,
                        const float* __restrict__ a_src,
                        const float* __restrict__ a_dst,
                        float* __restrict__ e_src, float* __restrict__ e_dst, int N)
{
  const int n    = (int)((blockIdx.x * (size_t)blockDim.x + threadIdx.x) >> 5);
  const int lane = threadIdx.x & 31;
  if (n >= N) return;
  const float* hp = hx + (size_t)n * GAT_HID + lane * 8;
  float ss = 0.f, sd = 0.f;
#pragma unroll
  for (int i = 0; i < 8; ++i) {
    const float v = hp[i];
    ss += v * a_src[lane * 8 + i];
    sd += v * a_dst[lane * 8 + i];
  }
  // reduce within each 8-lane group (one head per group) — wave32 shuffles
#pragma unroll
  for (int off = 4; off >= 1; off >>= 1) {
    ss += __shfl_xor(ss, off, 32);
    sd += __shfl_xor(sd, off, 32);
  }
  if ((lane & 7) == 0) {
    const int h = lane >> 3;
    e_src[(size_t)n * GAT_HEADS + h] = ss;
    e_dst[(size_t)n * GAT_HEADS + h] = sd;
  }
}

// ---------------------------------------------------------------------------
// Edge pass 1: segment max of leaky_relu(e_src[s]+e_dst[d]) per (dst, head)
// via atomicMax on order-preserving u32 (L2-resident atomics).
// ---------------------------------------------------------------------------
__global__ __launch_bounds__(256)
void edge_max_kernel(const int* __restrict__ ei,
                     const float* __restrict__ es, const float* __restrict__ ed,
                     unsigned* __restrict__ emax, int E, int Etot)
{
  const int idx = blockIdx.x * blockDim.x + threadIdx.x;
  if (idx >= Etot * GAT_HEADS) return;
  const int e = idx >> 2, h = idx & 3;
  int s, d; edge_sd(ei, E, e, s, d);
  float v = es[(size_t)s * GAT_HEADS + h] + ed[(size_t)d * GAT_HEADS + h];
  v = (v >= 0.f) ? v : NEG_SLOPE * v;
  atomicMax(emax + (size_t)d * GAT_HEADS + h, f32_flip(v));
}

// ---------------------------------------------------------------------------
// Edge pass 2: denom[d,h] += exp(e - emax[d,h])
// ---------------------------------------------------------------------------
__global__ __launch_bounds__(256)
void edge_denom_kernel(const int* __restrict__ ei,
                       const float* __restrict__ es, const float* __restrict__ ed,
                       const unsigned* __restrict__ emax, float* __restrict__ denom,
                       int E, int Etot)
{
  const int idx = blockIdx.x * blockDim.x + threadIdx.x;
  if (idx >= Etot * GAT_HEADS) return;
  const int e = idx >> 2, h = idx & 3;
  int s, d; edge_sd(ei, E, e, s, d);
  float v = es[(size_t)s * GAT_HEADS + h] + ed[(size_t)d * GAT_HEADS + h];
  v = (v >= 0.f) ? v : NEG_SLOPE * v;
  const float m = f32_unflip(emax[(size_t)d * GAT_HEADS + h]);
  atomicAdd(denom + (size_t)d * GAT_HEADS + h, __expf(v - m));
}

// ---------------------------------------------------------------------------
// Edge pass 3: agg[d,:] += hx[s,:] * alpha   (wave per edge, 8 channels/lane)
// ---------------------------------------------------------------------------
__global__ __launch_bounds__(256)
void edge_agg_kernel(const int* __restrict__ ei,
                     const float* __restrict__ es, const float* __restrict__ ed,
                     const unsigned* __restrict__ emax, const float* __restrict__ denom,
                     const float* __restrict__ hx, float* __restrict__ agg,
                     int E, int Etot)
{
  const int e    = (int)((blockIdx.x * (size_t)blockDim.x + threadIdx.x) >> 5);
  const int lane = threadIdx.x & 31;
  if (e >= Etot) return;
  int s, d; edge_sd(ei, E, e, s, d);
  const int h = lane >> 3;                       // channel block -> head
  float v = es[(size_t)s * GAT_HEADS + h] + ed[(size_t)d * GAT_HEADS + h];
  v = (v >= 0.f) ? v : NEG_SLOPE * v;
  const float m = f32_unflip(emax[(size_t)d * GAT_HEADS + h]);
  const float alpha = __expf(v - m) / denom[(size_t)d * GAT_HEADS + h];
  const float* hp = hx  + (size_t)s * GAT_HID + lane * 8;
  float*       op = agg + (size_t)d * GAT_HID + lane * 8;
#pragma unroll
  for (int i = 0; i < 8; ++i)
    atomicAdd(op + i, hp[i] * alpha);
}

// ---------------------------------------------------------------------------
// y = ELU(LayerNorm(agg + bias + residual)); wave per node, wave32 reduction.
// Safe to alias residual == out (wave reads all of its node before writing).
// ---------------------------------------------------------------------------
__global__ __launch_bounds__(256)
void ln_elu_kernel(const float* __restrict__ agg, const float* __restrict__ bias,
                   const float* __restrict__ residual,
                   const float* __restrict__ g, const float* __restrict__ be,
                   float* __restrict__ out, int N)
{
  const int n    = (int)((blockIdx.x * (size_t)blockDim.x + threadIdx.x) >> 5);
  const int lane = threadIdx.x & 31;
  if (n >= N) return;
  float v[8];
  float s1 = 0.f, s2 = 0.f;
#pragma unroll
  for (int i = 0; i < 8; ++i) {
    const int c = lane * 8 + i;
    v[i] = agg[(size_t)n * GAT_HID + c] + bias[c] + residual[(size_t)n * GAT_HID + c];
    s1 += v[i];
    s2 += v[i] * v[i];
  }
#pragma unroll
  for (int off = 16; off >= 1; off >>= 1) {
    s1 += __shfl_xor(s1, off, 32);
    s2 += __shfl_xor(s2, off, 32);
  }
  const float mu   = s1 * (1.0f / GAT_HID);
  const float var  = s2 * (1.0f / GAT_HID) - mu * mu;
  const float rstd = rsqrtf(var + 1e-5f);
#pragma unroll
  for (int i = 0; i < 8; ++i) {
    const int c = lane * 8 + i;
    const float y = (v[i] - mu) * rstd * g[c] + be[c];
    out[(size_t)n * GAT_HID + c] = (y > 0.f) ? y : (__expf(y) - 1.f);
  }
}

// ---------------------------------------------------------------------------
extern "C" void kernel_launch(void* const* d_in, const int* in_sizes, int n_in,
                              void* d_out, int out_size, void* d_ws, size_t ws_size,
                              hipStream_t stream) {
  const float* x   = (const float*)d_in[0];
  const int*   ei  = (const int*)  d_in[1];
  const float* W1  = (const float*)d_in[2];
  const float* as1 = (const float*)d_in[3];
  const float* ad1 = (const float*)d_in[4];
  const float* b1  = (const float*)d_in[5];
  const float* W2  = (const float*)d_in[6];
  const float* as2 = (const float*)d_in[7];
  const float* ad2 = (const float*)d_in[8];
  const float* b2  = (const float*)d_in[9];
  const float* g1  = (const float*)d_in[10];
  const float* be1 = (const float*)d_in[11];
  const float* g2  = (const float*)d_in[12];
  const float* be2 = (const float*)d_in[13];
  const float* Wp  = (const float*)d_in[14];
  const float* bp  = (const float*)d_in[15];

  const int N    = in_sizes[0] / GAT_IN;   // 30000
  const int E    = in_sizes[1] / 2;        // 480000
  const int Etot = E + N;                  // edges + self-loops

  // ws layout (floats): [agg N*256][emax N*4][denom N*4][esrc N*4][edst N*4]
  //                     [resid N*256][hx N*256]
  float* ws = (float*)d_ws;
  size_t off = 0;
  float*    agg   = ws + off;              off += (size_t)N * GAT_HID;
  unsigned* emax  = (unsigned*)(ws + off); off += (size_t)N * GAT_HEADS;
  float*    denom = ws + off;              off += (size_t)N * GAT_HEADS;
  float*    esrc  = ws + off;              off += (size_t)N * GAT_HEADS;
  float*    edst  = ws + off;              off += (size_t)N * GAT_HEADS;
  float*    resid = ws + off;              off += (size_t)N * GAT_HID;
  float*    hx    = ws + off;              off += (size_t)N * GAT_HID;
  float*    h1    = (float*)d_out;         // layer-1 output lives in d_out

  const int  zeroCnt    = N * GAT_HID + 2 * N * GAT_HEADS;  // agg+emax+denom
  const dim3 blk(256);
  const dim3 gZero((zeroCnt + 255) / 256);
  const dim3 gGemm(((N + 15) / 16 + 7) / 8, GAT_HID / 64);
  const dim3 gNode((N + 7) / 8);                      // wave per node
  const dim3 gEdgeH((Etot * GAT_HEADS + 255) / 256);  // thread per (edge,head)
  const dim3 gEdgeW((Etot + 7) / 8);                  // wave per edge

  // residual projection (shared by layer 1)
  wmma_gemm_kernel<GAT_IN><<<gGemm, blk, 0, stream>>>(x, Wp, bp, resid, N);

  // ----- layer 1 -----
  zero_u32_kernel<<<gZero, blk, 0, stream>>>((unsigned*)agg, zeroCnt);
  wmma_gemm_kernel<GAT_IN><<<gGemm, blk, 0, stream>>>(x, W1, nullptr, hx, N);
  attn_logits_kernel<<<gNode, blk, 0, stream>>>(hx, as1, ad1, esrc, edst, N);
  edge_max_kernel  <<<gEdgeH, blk, 0, stream>>>(ei, esrc, edst, emax, E, Etot);
  edge_denom_kernel<<<gEdgeH, blk, 0, stream>>>(ei, esrc, edst, emax, denom, E, Etot);
  edge_agg_kernel  <<<gEdgeW, blk, 0, stream>>>(ei, esrc, edst, emax, denom, hx, agg, E, Etot);
  ln_elu_kernel    <<<gNode, blk, 0, stream>>>(agg, b1, resid, g1, be1, h1, N);

  // ----- layer 2 (residual = h1, held in d_out; final write is in place) -----
  zero_u32_kernel<<<gZero, blk, 0, stream>>>((unsigned*)agg, zeroCnt);
  wmma_gemm_kernel<GAT_HID><<<gGemm, blk, 0, stream>>>(h1, W2, nullptr, hx, N);
  attn_logits_kernel<<<gNode, blk, 0, stream>>>(hx, as2, ad2, esrc, edst, N);
  edge_max_kernel  <<<gEdgeH, blk, 0, stream>>>(ei, esrc, edst, emax, E, Etot);
  edge_denom_kernel<<<gEdgeH, blk, 0, stream>>>(ei, esrc, edst, emax, denom, E, Etot);
  edge_agg_kernel  <<<gEdgeW, blk, 0, stream>>>(ei, esrc, edst, emax, denom, hx, agg, E, Etot);
  ln_elu_kernel    <<<gNode, blk, 0, stream>>>(agg, b2, h1, g2, be2, (float*)d_out, N);
}